// HashGridEncode_26353919328262
// MI455X (gfx1250) — compile-verified
//
#include <hip/hip_runtime.h>
#include <hip/hip_bf16.h>
#include <stdint.h>

// HashGridEncode (Instant-NGP multires hash encoding), MI455X / gfx1250.
//
// Roofline: ~1 GFLOP vs ~140 MB HBM streams + 1 GiB of random 8B gathers.
// The 64 MB embedding table fits in the 192 MB L2 -> gathers are L2-resident
// *if* we don't let the 128 MB write-once output stream evict it. So:
//  - output stores use the CDNA5 non-temporal store hint (TH=NT) to keep the
//    table pinned in L2 while the output streams through to HBM,
//  - level-fast thread mapping gives perfectly coalesced b64 output stores,
//  - all 8 gather loads per item issue back-to-back (one loadcnt wait),
//  - T = 2^19 is pow2 -> hash modulo is an AND mask (urem fallback is
//    execz-skipped, kept only for robustness),
//  - global_prefetch_b8 warms the 12 MB xyz stream.
// WMMA is inapplicable: the 8-corner weighted sum has no shared operand
// across a 16x16 tile (both einsum operands vary per point).

#define HG_L 16   // levels (fixed by reference: L = 16, power of two)

typedef float f32x2 __attribute__((ext_vector_type(2)));

__global__ __launch_bounds__(256)
void HashGridEncode_26353919328262_kernel(const float*  __restrict__ xyz,
                                          const f32x2*  __restrict__ emb,
                                          const int*    __restrict__ offsets,
                                          const int*    __restrict__ resolutions,
                                          const float*  __restrict__ min_xyz,
                                          const float*  __restrict__ max_xyz,
                                          f32x2*        __restrict__ out,
                                          int B)
{
    const int stride = gridDim.x * blockDim.x;
    const int total  = B * HG_L;

    // Uniform per-dispatch constants (scalar loads; division hoisted).
    const float mn0 = min_xyz[0], mn1 = min_xyz[1], mn2 = min_xyz[2];
    const float inv0 = 1.0f / (max_xyz[0] - mn0);
    const float inv1 = 1.0f / (max_xyz[1] - mn1);
    const float inv2 = 1.0f / (max_xyz[2] - mn2);

    for (int t = blockIdx.x * blockDim.x + threadIdx.x; t < total; t += stride)
    {
        const int p = t >> 4;          // point index (HG_L == 16)
        const int l = t & (HG_L - 1);  // level index (fast dim -> coalesced out)

        // gfx1250 global_prefetch_b8: warm next grid-stride iteration's xyz.
        {
            const int tn = t + stride;
            if ((tn < total) && ((t & (HG_L - 1)) == 0))
                __builtin_prefetch(xyz + (size_t)(tn >> 4) * 3, 0, 1);
        }

        const float x = xyz[p * 3 + 0];
        const float y = xyz[p * 3 + 1];
        const float z = xyz[p * 3 + 2];

        const float ux = (x - mn0) * inv0;
        const float uy = (y - mn1) * inv1;
        const float uz = (z - mn2) * inv2;
        const bool valid = (ux >= 0.f) & (ux <= 1.f) &
                           (uy >= 0.f) & (uy <= 1.f) &
                           (uz >= 0.f) & (uz <= 1.f);

        const float rf  = (float)resolutions[l];
        const float rm1 = rf - 1.0f;

        const float px = ux * rf, py = uy * rf, pz = uz * rf;
        const float p0x = fminf(fmaxf(floorf(px), 0.f), rm1);
        const float p0y = fminf(fmaxf(floorf(py), 0.f), rm1);
        const float p0z = fminf(fmaxf(floorf(pz), 0.f), rm1);
        const float fx = px - p0x, fy = py - p0y, fz = pz - p0z;

        const uint32_t ix = (uint32_t)p0x;
        const uint32_t iy = (uint32_t)p0y;
        const uint32_t iz = (uint32_t)p0z;

        const uint32_t o0   = (uint32_t)offsets[l];
        const uint32_t size = (uint32_t)offsets[l + 1] - o0;
        const uint32_t mask = size - 1u;

        // Hash factors: corner hash = hx[bx] ^ hy[by] ^ hz[bz]  (2 mults total)
        const uint32_t hx[2] = { ix,               ix + 1u };
        const uint32_t hy[2] = { iy * 2654435761u, iy * 2654435761u + 2654435761u };
        const uint32_t hz[2] = { iz * 805459861u,  iz * 805459861u  + 805459861u };

        uint32_t idxs[8];
        if ((size & mask) == 0u) {
            // Fast path: power-of-two table (reference: T = 2^19) -> AND mask.
            #pragma unroll
            for (int c = 0; c < 8; ++c) {
                const uint32_t h = hx[c & 1] ^ hy[(c >> 1) & 1] ^ hz[(c >> 2) & 1];
                idxs[c] = (h & mask) + o0;
            }
        } else {
            // Generic fallback (never taken here; execz-skipped at runtime).
            #pragma unroll
            for (int c = 0; c < 8; ++c) {
                const uint32_t h = hx[c & 1] ^ hy[(c >> 1) & 1] ^ hz[(c >> 2) & 1];
                idxs[c] = (h % size) + o0;
            }
        }

        // Issue all 8 independent b64 gathers before consuming any (one
        // loadcnt wait covers the batch; hits land in the 192 MB L2).
        f32x2 g[8];
        #pragma unroll
        for (int c = 0; c < 8; ++c) g[c] = emb[idxs[c]];

        const float wx[2] = { 1.f - fx, fx };
        const float wy[2] = { 1.f - fy, fy };
        const float wz[2] = { 1.f - fz, fz };

        float acc0 = 0.f, acc1 = 0.f;
        #pragma unroll
        for (int c = 0; c < 8; ++c) {
            const float w = wx[c & 1] * wy[(c >> 1) & 1] * wz[(c >> 2) & 1];
            acc0 = fmaf(w, g[c].x, acc0);
            acc1 = fmaf(w, g[c].y, acc1);
        }

        const float vm = valid ? 1.0f : 0.0f;
        f32x2 r; r.x = acc0 * vm; r.y = acc1 * vm;
        // t = p*16 + l -> floats [p*32 + 2l, p*32 + 2l + 1] == out[b, l*F + f].
        // Non-temporal: write-once 128 MB stream must not evict the embedding
        // table from L2. Consecutive lanes -> consecutive 8B -> coalesced.
        __builtin_nontemporal_store(r, &out[t]);
    }
}

extern "C" void kernel_launch(void* const* d_in, const int* in_sizes, int n_in,
                              void* d_out, int out_size, void* d_ws, size_t ws_size,
                              hipStream_t stream) {
    (void)n_in; (void)out_size; (void)d_ws; (void)ws_size;

    const float* xyz         = (const float*)d_in[0];   // (B, 3) f32
    const f32x2* emb         = (const f32x2*)d_in[1];   // (16*T, 2) f32
    const int*   offsets     = (const int*)d_in[2];     // (17,) i32
    const int*   resolutions = (const int*)d_in[3];     // (16,) i32
    const float* mn          = (const float*)d_in[4];   // (3,) f32
    const float* mx          = (const float*)d_in[5];   // (3,) f32
    f32x2*       out         = (f32x2*)d_out;           // (B, 32) f32

    const int B     = in_sizes[0] / 3;
    const int total = B * HG_L;
    const int block = 256;                 // 8 wave32s per block
    int blocks = (total + block - 1) / block;
    if (blocks > 16384) blocks = 16384;    // grid-stride: ~4 items/thread

    HashGridEncode_26353919328262_kernel<<<blocks, block, 0, stream>>>(
        xyz, emb, offsets, resolutions, mn, mx, out, B);
}